// MoDRouter_34428457844853
// MI455X (gfx1250) — compile-verified
//
#include <hip/hip_runtime.h>
#include <math.h>

typedef __attribute__((ext_vector_type(2))) float v2f;
typedef __attribute__((ext_vector_type(8))) float v8f;

#define BB 8
#define SS 4096
#define DD 1024
#define KTOP 3072

// ---- monotonic float<->uint mapping for sortable keys ----
__device__ __forceinline__ unsigned enc_f32(float f) {
    unsigned u = __float_as_uint(f);
    return (u & 0x80000000u) ? ~u : (u | 0x80000000u);
}
__device__ __forceinline__ float dec_f32(unsigned e) {
    unsigned u = (e & 0x80000000u) ? (e ^ 0x80000000u) : ~e;
    return __uint_as_float(u);
}

// ------------------------------------------------------------------
// K0: scores[b,s] = dot(x[b,s,:], w) via V_WMMA_F32_16X16X4_F32.
// One wave handles 16 tokens. B-matrix = w replicated into all 16
// columns, so every C column equals the 16 scores.
// A layout (32-bit 16x4): lanes 0-15 M=lane, {v0,v1}={K0,K1};
//                         lanes 16-31 M=lane-16, {v0,v1}={K2,K3}.
// B layout (4x16): v0 = row K0 (lanes<16) / K2 (lanes>=16); v1 = K1/K3.
// C layout: lane<16 -> rows 0..7 in c[0..7]; lane>=16 -> rows 8..15.
// ------------------------------------------------------------------
__global__ void __launch_bounds__(256) mod_scores_kernel(
        const float* __restrict__ x,
        const float* __restrict__ w,
        float* __restrict__ scores) {
    __shared__ float wl[DD];
    for (int i = threadIdx.x; i < DD; i += 256) wl[i] = w[i];
    __syncthreads();

    const int lane = threadIdx.x & 31;
    const int wave = (blockIdx.x * 256 + threadIdx.x) >> 5;
    const int row0 = wave * 16;                     // 16 tokens per wave
    const int sub  = (lane < 16) ? lane : (lane - 16);
    const int koff = (lane < 16) ? 0 : 2;
    const float* xrow = x + (size_t)(row0 + sub) * DD;

    v8f c = {};
    for (int k = 0; k < DD; k += 4) {
        v2f a, b;
        a.x = xrow[k + koff];
        a.y = xrow[k + koff + 1];
        b.x = wl[k + koff];
        b.y = wl[k + koff + 1];
        c = __builtin_amdgcn_wmma_f32_16x16x4_f32(
                false, a, false, b, (short)0, c, false, false);
    }
    if (lane == 0) {
#pragma unroll
        for (int r = 0; r < 8; ++r) scores[row0 + r] = c[r];
    } else if (lane == 16) {
#pragma unroll
        for (int r = 0; r < 8; ++r) scores[row0 + 8 + r] = c[r];
    }
}

// ------------------------------------------------------------------
// K1: per batch row -- bitonic sort of 4096 (score,pos) keys in LDS,
// descending score / ascending pos; emit rankmap (pos -> rank or -1)
// and softmax weights over the top-K ranks.
// ------------------------------------------------------------------
__global__ void __launch_bounds__(1024) mod_sort_kernel(
        const float* __restrict__ scores,
        int* __restrict__ rankmap,
        float* __restrict__ weights) {
    __shared__ unsigned long long keys[SS];   // 32 KB
    __shared__ float red[1024];               //  4 KB

    const int b   = blockIdx.x;
    const int tid = threadIdx.x;
    const float* sc = scores + (size_t)b * SS;

    for (int i = tid; i < SS; i += 1024) {
        unsigned e = enc_f32(sc[i]);
        keys[i] = ((unsigned long long)e << 32) | (unsigned)(SS - 1 - i);
    }
    __syncthreads();

    // bitonic sort, descending
    for (int k = 2; k <= SS; k <<= 1) {
        for (int j = k >> 1; j > 0; j >>= 1) {
            for (int t = tid; t < SS; t += 1024) {
                int ixj = t ^ j;
                if (ixj > t) {
                    unsigned long long a0 = keys[t];
                    unsigned long long a1 = keys[ixj];
                    bool up = ((t & k) == 0);
                    bool swap = up ? (a0 < a1) : (a0 > a1);
                    if (swap) { keys[t] = a1; keys[ixj] = a0; }
                }
            }
            __syncthreads();
        }
    }

    // inverse map: every position has a rank; -1 if not routed
    for (int r = tid; r < SS; r += 1024) {
        int pos = (SS - 1) - (int)(unsigned)(keys[r] & 0xFFFFFFFFu);
        rankmap[(size_t)b * SS + pos] = (r < KTOP) ? r : -1;
    }

    // softmax over ranks [0, KTOP)
    float maxs = dec_f32((unsigned)(keys[0] >> 32));
    float lsum = 0.0f;
    for (int r = tid; r < KTOP; r += 1024) {
        float s = dec_f32((unsigned)(keys[r] >> 32));
        lsum += expf(s - maxs);
    }
    red[tid] = lsum;
    __syncthreads();
    for (int off = 512; off > 0; off >>= 1) {
        if (tid < off) red[tid] += red[tid + off];
        __syncthreads();
    }
    const float inv = 1.0f / red[0];
    for (int r = tid; r < KTOP; r += 1024) {
        float s = dec_f32((unsigned)(keys[r] >> 32));
        weights[(size_t)b * KTOP + r] = expf(s - maxs) * inv;
    }
}

// ------------------------------------------------------------------
// K2: streaming blend. One block per token row, float4 per thread.
// out = (rank>=0) ? w*processed[b,rank] + (1-w)*x : x
// ------------------------------------------------------------------
__global__ void __launch_bounds__(256) mod_blend_kernel(
        const float* __restrict__ x,
        const float* __restrict__ processed,
        const int* __restrict__ rankmap,
        const float* __restrict__ weights,
        float* __restrict__ out) {
    const int row = blockIdx.x;          // 0 .. B*S-1
    const int b   = row >> 12;           // row / S
    const int r   = rankmap[row];
    const int t   = threadIdx.x;         // 256 threads * float4 = 1024 floats

    const float4* xr = (const float4*)(x + (size_t)row * DD);
    float4*       po = (float4*)(out + (size_t)row * DD);
    float4 xv = xr[t];

    if (r >= 0) {
        float wgt = weights[(size_t)b * KTOP + r];
        float om  = 1.0f - wgt;
        const float4* pr =
            (const float4*)(processed + ((size_t)b * KTOP + r) * DD);
        float4 pv = pr[t];
        xv.x = pv.x * wgt + om * xv.x;
        xv.y = pv.y * wgt + om * xv.y;
        xv.z = pv.z * wgt + om * xv.z;
        xv.w = pv.w * wgt + om * xv.w;
    }
    po[t] = xv;
}

extern "C" void kernel_launch(void* const* d_in, const int* in_sizes, int n_in,
                              void* d_out, int out_size, void* d_ws, size_t ws_size,
                              hipStream_t stream) {
    const float* x         = (const float*)d_in[0];   // (B,S,D)
    const float* processed = (const float*)d_in[1];   // (B,K,D)
    const float* w_router  = (const float*)d_in[2];   // (D,)
    float* out = (float*)d_out;

    // workspace layout
    char* ws = (char*)d_ws;
    float* scores  = (float*)(ws);                         // B*S floats  (128 KB)
    int*   rankmap = (int*)(ws + (size_t)BB * SS * 4);     // B*S ints    (128 KB)
    float* weights = (float*)(ws + (size_t)BB * SS * 8);   // B*K floats  ( 96 KB)

    // K0: 2048 waves total, 8 waves (256 thr) per block -> 256 blocks
    mod_scores_kernel<<<(BB * SS) / (16 * 8), 256, 0, stream>>>(x, w_router, scores);

    // K1: one block per batch row
    mod_sort_kernel<<<BB, 1024, 0, stream>>>(scores, rankmap, weights);

    // K2: one block per token row
    mod_blend_kernel<<<BB * SS, 256, 0, stream>>>(x, processed, rankmap, weights, out);

    (void)in_sizes; (void)n_in; (void)out_size; (void)ws_size;
}